// Attn_head_40364102648200
// MI455X (gfx1250) — compile-verified
//
#include <hip/hip_runtime.h>
#include <hip/hip_bf16.h>
#include <math.h>

// CDNA5 wave32 WMMA fragment types
typedef __attribute__((ext_vector_type(2))) float v2f;   // f32 A/B frag: 16x4 / 4x16 = 64 f32 / 32 lanes
typedef __attribute__((ext_vector_type(8))) float v8f;   // f32 C/D frag: 16x16 = 256 f32 / 32 lanes

#define B_DIM   4
#define C_IN    512
#define N_DIM   4096
#define OUT_DIM 256
#define M_PAD   16   // pad batch dim 4 -> 16 for the 16x16 WMMA tile

// ---------------------------------------------------------------------------
// Stage 1: colsum[b][c] = sum_n x[b,c,0,n]; writes zero-padded P[16][512].
// One block per padded row (16*512 = 8192 blocks); rows >= 2048 are padding.
// HBM-bound: streams 32 MB of x.
// ---------------------------------------------------------------------------
__global__ __launch_bounds__(256) void rowsum_kernel(const float* __restrict__ x,
                                                     float* __restrict__ P) {
    __shared__ float red[256];
    const int row = blockIdx.x;                 // 0 .. 16*512-1
    const int t   = threadIdx.x;
    if (row >= B_DIM * C_IN) {                  // zero padding rows (b = 4..15)
        if (t == 0) P[row] = 0.0f;
        return;
    }
    const float* __restrict__ xr = x + (size_t)row * N_DIM;
    float s = 0.0f;
#pragma unroll
    for (int i = 0; i < 4; ++i) {               // 4 x float4 per thread = 4096 floats/block
        const float4 v = *(const float4*)(xr + (size_t)(i * 256 + t) * 4);
        s += v.x + v.y + v.z + v.w;
    }
    red[t] = s;
    __syncthreads();
#pragma unroll
    for (int off = 128; off > 0; off >>= 1) {
        if (t < off) red[t] += red[t + off];
        __syncthreads();
    }
    if (t == 0) P[row] = red[0];
}

// ---------------------------------------------------------------------------
// Stage 2: S[b][o] = sum_c P[b][c] * W1[o][c], then ELU.
// Single workgroup, 16 waves; wave w computes the 16x16 tile covering
// o = 16w..16w+15, accumulating K=512 in 128 V_WMMA_F32_16X16X4_F32 steps.
// A frag (16x4 f32): lanes 0-15 & 16-31 both hold M = lane&15;
//   VGPR0 -> K = 2*half + 0, VGPR1 -> K = 2*half + 1  (half = lane>>4).
// B frag (4x16 f32): lane holds column N = lane&15, same K split.
// ---------------------------------------------------------------------------
__global__ __launch_bounds__(512) void gemm_s_kernel(const float* __restrict__ P,   // [16][512]
                                                     const float* __restrict__ W1,  // [256][512]
                                                     float* __restrict__ S) {       // [4][256] (elu'd)
    const int lane  = threadIdx.x & 31;
    const int wave  = threadIdx.x >> 5;     // 0..15 -> N tile
    const int half  = lane >> 4;            // 0 or 1
    const int l15   = lane & 15;
    const int obase = wave * 16;

    const float* __restrict__ Arow = P  + (size_t)l15 * C_IN          + 2 * half;
    const float* __restrict__ Brow = W1 + (size_t)(obase + l15) * C_IN + 2 * half;

    v8f acc = {};
#pragma unroll 8
    for (int k = 0; k < C_IN; k += 4) {
        v2f a, b;
        a.x = Arow[k];
        a.y = Arow[k + 1];
        b.x = Brow[k];
        b.y = Brow[k + 1];
        // D = A*B + C  (fp32 WMMA, exact vs reference accumulation precision)
        acc = __builtin_amdgcn_wmma_f32_16x16x4_f32(
            /*neg_a=*/false, a, /*neg_b=*/false, b,
            /*c_mod=*/(short)0, acc, /*reuse_a=*/false, /*reuse_b=*/false);
    }

    // D layout: VGPR r -> M=r (lanes 0-15), M=r+8 (lanes 16-31); N = lane&15.
    // Only batch rows 0..3 are real -> acc[0..3] on lanes 0..15.
    if (half == 0) {
#pragma unroll
        for (int b = 0; b < B_DIM; ++b) {
            float v = acc[b];
            v = (v > 0.0f) ? v : expm1f(v);          // ELU(alpha=1)
            S[b * OUT_DIM + obase + l15] = v;
        }
    }
}

// ---------------------------------------------------------------------------
// Stage 3: out[b][o][0][n] = S_elu[b][o] for all n.  16 MB broadcast write,
// float4 stores, fully coalesced.
// ---------------------------------------------------------------------------
__global__ __launch_bounds__(256) void bcast_kernel(const float* __restrict__ S,
                                                    float* __restrict__ out) {
    const int idx = blockIdx.x * blockDim.x + threadIdx.x;  // 1M threads
    const int e   = idx << 2;                               // element index (x4)
    const int bo  = e >> 12;                                // N_DIM = 4096 per (b,o) row
    const float v = S[bo];
    float4 val;
    val.x = v; val.y = v; val.z = v; val.w = v;
    *(float4*)(out + e) = val;
}

extern "C" void kernel_launch(void* const* d_in, const int* in_sizes, int n_in,
                              void* d_out, int out_size, void* d_ws, size_t ws_size,
                              hipStream_t stream) {
    const float* x  = (const float*)d_in[0];   // [4,512,1,4096]
    const float* W1 = (const float*)d_in[1];   // [256,512]
    // d_in[2] (w2) and d_in[3] (bias_mat) are mathematically dead:
    // softmax over the singleton axis=1 makes coefs == 1 regardless of f.
    float* out = (float*)d_out;                // [4,256,1,4096]

    float* P = (float*)d_ws;                   // [16][512] padded colsums (32 KB)
    float* S = P + M_PAD * C_IN;               // [4][256]  elu'd S        (4 KB)

    rowsum_kernel<<<M_PAD * C_IN, 256, 0, stream>>>(x, P);
    gemm_s_kernel<<<1, 512, 0, stream>>>(P, W1, S);
    const int total = B_DIM * OUT_DIM * N_DIM;               // 4,194,304 floats
    bcast_kernel<<<total / (256 * 4), 256, 0, stream>>>(S, out);
}